// gPool_20658792694321
// MI455X (gfx1250) — compile-verified
//
#include <hip/hip_runtime.h>
#include <math.h>

// ---------------- problem constants (match reference) ----------------
#define B_G     64
#define N_PER   4096
#define N_NODES (B_G * N_PER)        // 262144
#define C_DIM   128
#define E_EDGES 4194304
#define K_KEEP  3277                 // ceil(0.8 * 4096)
#define NK      (B_G * K_KEEP)       // 209728

// ---------------- output layout (floats, concatenated in return order) ----
#define O_XP    0
#define XP_SZ   ((size_t)NK * C_DIM)            // 26,845,184
#define O_E0    (XP_SZ)                          // new_edge_index row 0
#define O_E1    (O_E0 + (size_t)E_EDGES)         // new_edge_index row 1
#define O_BATCH (O_E0 + 2 * (size_t)E_EDGES)
#define O_PERM  (O_BATCH + (size_t)NK)

// ---------------- workspace layout (bytes) ----------------
#define WS_SCORE   0
#define WS_NODEMAP ((size_t)N_NODES * 4)
#define WS_PERM    (WS_NODEMAP + (size_t)N_NODES * 4)
#define WS_OFF     (WS_PERM + (size_t)NK * 4)    // 1025 ints (counts -> exclusive offsets, [1024] = total)

typedef float v2f __attribute__((ext_vector_type(2)));
typedef float v8f __attribute__((ext_vector_type(8)));

// =====================================================================
// K1: score = x @ coefficient via V_WMMA_F32_16X16X4_F32
// One wave per 16-node tile; tile staged in LDS (padded, conflict-free).
// B-matrix is the coefficient broadcast across all 16 columns using the
// same (VGPR, lane-half) -> K mapping as the documented A layout.
// =====================================================================
#define SCORE_WAVES 4
#define PAD 132
__global__ __launch_bounds__(SCORE_WAVES * 32) void k_score(
    const float* __restrict__ x, const float* __restrict__ coef,
    float* __restrict__ score) {
  __shared__ float tile[SCORE_WAVES][16][PAD];
  const int tid  = threadIdx.x;
  const int wid  = tid >> 5;
  const int lane = tid & 31;
  const int t    = blockIdx.x * SCORE_WAVES + wid;   // tile id
  const int base = t * 16;                           // first node of tile

  // stage 16 rows x 128 cols, coalesced float4 per lane
  const float4* xr = (const float4*)(x + (size_t)base * C_DIM);
#pragma unroll
  for (int i = 0; i < 16; ++i) {
    float4 v = xr[i * 32 + lane];
    *(float4*)&tile[wid][i][lane * 4] = v;
  }
  __syncthreads();

  // A fragment: lane L holds row M = L&15; VGPR0 -> K = k0 + (L<16?0:2),
  // VGPR1 -> K = +1 (documented 32-bit 16x4 A layout). B mirrors it.
  const int off = (lane < 16) ? 0 : 2;
  const float* row = &tile[wid][lane & 15][0];

  v8f acc = {0.f, 0.f, 0.f, 0.f, 0.f, 0.f, 0.f, 0.f};
#pragma unroll
  for (int k0 = 0; k0 < C_DIM; k0 += 4) {
    v2f a, b;
    a.x = row[k0 + off];
    a.y = row[k0 + off + 1];
    b.x = coef[k0 + off];
    b.y = coef[k0 + off + 1];
    acc = __builtin_amdgcn_wmma_f32_16x16x4_f32(
        false, a, false, b, (short)0, acc, false, false);
  }

  // broadcast-B => every D column identical. lanes 0-7 cover M=0..7,
  // lanes 16-23 cover M=8..15.
  if (((lane >> 3) & 1) == 0) {
    float v = 0.f;
#pragma unroll
    for (int q = 0; q < 8; ++q)
      if ((lane & 7) == q) v = acc[q];
    int m = (lane & 7) + ((lane >> 4) << 3);
    score[base + m] = v;
  }
}

// =====================================================================
// K2: per-graph bitonic sort (descending score, ascending idx tiebreak),
// emit perm / batch outputs, perm_i, and node_map (kept rank or -1).
// =====================================================================
__global__ __launch_bounds__(1024) void k_topk(
    const float* __restrict__ score, int* __restrict__ perm_i,
    int* __restrict__ node_map, float* __restrict__ out) {
  __shared__ float sk[N_PER];
  __shared__ int   si[N_PER];
  const int g = blockIdx.x, tid = threadIdx.x;

  for (int i = tid; i < N_PER; i += 1024) {
    sk[i] = score[g * N_PER + i];
    si[i] = i;
  }
  __syncthreads();

  for (int k = 2; k <= N_PER; k <<= 1) {
    for (int j = k >> 1; j > 0; j >>= 1) {
      for (int i = tid; i < N_PER; i += 1024) {
        int ixj = i ^ j;
        if (ixj > i) {
          float s1 = sk[i], s2 = sk[ixj];
          int   i1 = si[i], i2 = si[ixj];
          bool firstRanks = (s1 > s2) || (s1 == s2 && i1 < i2);
          bool up = ((i & k) == 0);
          bool doSwap = up ? (!firstRanks) : firstRanks;  // descending overall
          if (doSwap) {
            sk[i] = s2; sk[ixj] = s1;
            si[i] = i2; si[ixj] = i1;
          }
        }
      }
      __syncthreads();
    }
  }

  for (int i = tid; i < N_PER; i += 1024) {
    int gi = g * N_PER + si[i];
    if (i < K_KEEP) {
      int pooled = g * K_KEEP + i;
      perm_i[pooled]        = gi;
      node_map[gi]          = pooled;
      out[O_PERM + pooled]  = (float)gi;
      out[O_BATCH + pooled] = (float)g;
    } else {
      node_map[gi] = -1;
    }
  }
}

// =====================================================================
// K3: x_pooled = x[perm] * sigmoid(score[perm]); one wave per row
// (32 lanes x float4 = 128 channels, fully coalesced).
// =====================================================================
__global__ __launch_bounds__(256) void k_gather(
    const float* __restrict__ x, const float* __restrict__ score,
    const int* __restrict__ perm_i, float* __restrict__ out) {
  const int tid = threadIdx.x, wid = tid >> 5, lane = tid & 31;
  const int row = blockIdx.x * 8 + wid;
  if (row >= NK) return;
  const int p = perm_i[row];
  const float s = score[p];
  const float gate = 1.0f / (1.0f + expf(-s));
  const float4* xr = (const float4*)(x + (size_t)p * C_DIM);
  float4 v = xr[lane];
  v.x *= gate; v.y *= gate; v.z *= gate; v.w *= gate;
  ((float4*)(out + O_XP + (size_t)row * C_DIM))[lane] = v;
}

// =====================================================================
// K4: per-block (4096 edges) valid-edge counts.
// =====================================================================
__global__ __launch_bounds__(256) void k_count(
    const int* __restrict__ ei, const int* __restrict__ node_map,
    int* __restrict__ counts) {
  __shared__ int wsum[8];
  const int tid = threadIdx.x, blk = blockIdx.x;
  const int lane = tid & 31, wid = tid >> 5;
  const int base = blk * 4096;
  int cnt = 0;
#pragma unroll
  for (int i = 0; i < 16; ++i) {
    int e = base + i * 256 + tid;
    int r = node_map[ei[e]];
    int c = node_map[ei[E_EDGES + e]];
    cnt += (r >= 0 && c >= 0) ? 1 : 0;
  }
  for (int d = 16; d > 0; d >>= 1) cnt += __shfl_down(cnt, d, 32);
  if (lane == 0) wsum[wid] = cnt;
  __syncthreads();
  if (tid == 0) {
    int s = 0;
#pragma unroll
    for (int w = 0; w < 8; ++w) s += wsum[w];
    counts[blk] = s;
  }
}

// =====================================================================
// K5: exclusive scan of the 1024 block counts (single block); [1024]=total.
// =====================================================================
__global__ __launch_bounds__(1024) void k_scan(int* __restrict__ off) {
  __shared__ int tmp[1024];
  const int tid = threadIdx.x;
  const int my = off[tid];
  tmp[tid] = my;
  __syncthreads();
  for (int d = 1; d < 1024; d <<= 1) {
    int v = tmp[tid];
    int add = (tid >= d) ? tmp[tid - d] : 0;
    __syncthreads();
    tmp[tid] = v + add;
    __syncthreads();
  }
  int incl = tmp[tid];
  off[tid] = incl - my;                 // exclusive
  if (tid == 1023) off[1024] = incl;    // total valid T
}

// =====================================================================
// K6: stable partition scatter. Valid edges -> [0,T) in original order,
// invalid -> [T,E) in original order as -1/-1. Ranks via ballot+popcount.
// =====================================================================
__global__ __launch_bounds__(256) void k_scatter(
    const int* __restrict__ ei, const int* __restrict__ node_map,
    const int* __restrict__ off, float* __restrict__ out) {
  __shared__ int wsum[8];
  const int tid = threadIdx.x, blk = blockIdx.x;
  const int lane = tid & 31, wid = tid >> 5;
  const int blkOff = off[blk];
  const int T = off[1024];
  const int base = blk * 4096;
  int carry = 0;
  for (int i = 0; i < 16; ++i) {
    int e = base + i * 256 + tid;
    int r = node_map[ei[e]];
    int c = node_map[ei[E_EDGES + e]];
    bool f = (r >= 0 && c >= 0);
    unsigned mask = (unsigned)__ballot(f);
    int laneRank = __popc(mask & ((1u << lane) - 1u));
    if (lane == 0) wsum[wid] = __popc(mask);
    __syncthreads();
    int wpre = 0, tot = 0;
#pragma unroll
    for (int w = 0; w < 8; ++w) {
      int v = wsum[w];
      tot += v;
      if (w < wid) wpre += v;
    }
    int vb = blkOff + carry + wpre + laneRank;   // #valid before e globally
    int pos = f ? vb : (T + (e - vb));
    out[O_E0 + pos] = f ? (float)r : -1.0f;
    out[O_E1 + pos] = f ? (float)c : -1.0f;
    carry += tot;
    __syncthreads();
  }
}

// =====================================================================
extern "C" void kernel_launch(void* const* d_in, const int* in_sizes, int n_in,
                              void* d_out, int out_size, void* d_ws,
                              size_t ws_size, hipStream_t stream) {
  const float* x    = (const float*)d_in[0];
  const float* coef = (const float*)d_in[1];
  const int*   ei   = (const int*)d_in[2];
  float* out = (float*)d_out;
  char* ws = (char*)d_ws;

  float* score    = (float*)(ws + WS_SCORE);
  int*   node_map = (int*)(ws + WS_NODEMAP);
  int*   perm_i   = (int*)(ws + WS_PERM);
  int*   off      = (int*)(ws + WS_OFF);

  // 1) scores via WMMA: 16384 tiles of 16 nodes, 4 waves/block
  k_score<<<N_NODES / 16 / SCORE_WAVES, SCORE_WAVES * 32, 0, stream>>>(x, coef, score);
  // 2) per-graph top-K sort + node_map/perm/batch outputs
  k_topk<<<B_G, 1024, 0, stream>>>(score, perm_i, node_map, out);
  // 3) gather + sigmoid gate
  k_gather<<<(NK + 7) / 8, 256, 0, stream>>>(x, score, perm_i, out);
  // 4-6) stable edge partition
  k_count<<<E_EDGES / 4096, 256, 0, stream>>>(ei, node_map, off);
  k_scan<<<1, 1024, 0, stream>>>(off);
  k_scatter<<<E_EDGES / 4096, 256, 0, stream>>>(ei, node_map, off, out);
}